// GAMultivectorLayer_43834436223688
// MI455X (gfx1250) — compile-verified
//
#include <hip/hip_runtime.h>

// ---- problem constants (match reference) ----
#define B_DIM   8192
#define IN_DIM  256
#define OUT_DIM 256
#define MV      8

// ---- tiling ----
#define MT 128          // batch rows per block
#define NT 16           // output cols per block
#define JT 16           // j's per LDS stage
#define KT (JT * 3)     // 48 K-values per stage (3 comps per j)
#define LDPAD 52        // LDS row pitch in dwords (gcd(52,64)=4 -> conflict-free 16-row reads)

typedef __attribute__((ext_vector_type(2))) float v2f;
typedef __attribute__((ext_vector_type(4))) float v4f;
typedef __attribute__((ext_vector_type(8))) float v8f;

// Kernel 1: broadcast bias into every (b, i, mv) slot. float4 streaming writes.
__global__ __launch_bounds__(256) void ga_fill_bias(const float4* __restrict__ bias4,
                                                    float4* __restrict__ y4,
                                                    int total4) {
  int idx = blockIdx.x * blockDim.x + threadIdx.x;
  int stride = gridDim.x * blockDim.x;
  for (int q = idx; q < total4; q += stride) {
    // bias pattern repeats every OUT_DIM*MV/4 = 512 float4's
    y4[q] = bias4[q & 511];
  }
}

// Kernel 2: f32 WMMA GEMM for components 0 and 4.
__global__ __launch_bounds__(256) void ga_gemm(const float* __restrict__ x,
                                               const float* __restrict__ w,
                                               const float* __restrict__ bias,
                                               float* __restrict__ y) {
  // A-operand staging: rows = batch, K-contiguous packing (padded rows)
  __shared__ float XSa[MT][LDPAD];  // [x0,x1,x2] per j  -> for c0
  __shared__ float XSb[MT][LDPAD];  // [x1,-x2,x4] per j -> for c4
  // B-operand staging, stored transposed [n][K] so each lane reads a b64 pair
  __shared__ float WSa[NT][LDPAD];  // [w0,w1,w2] per j
  __shared__ float WSb[NT][LDPAD];  // [w2,w1,w0] per j

  const int t    = threadIdx.x;   // 0..255
  const int lane = t & 31;
  const int wv   = t >> 5;        // wave id 0..7 (wave32)
  const int bm   = blockIdx.x * MT;
  const int n0   = blockIdx.y * NT;

  const int mhalf = lane & 15;        // M (A) / N (B) index within tile
  const int koff  = (lane >> 4) * 2;  // lanes 16..31 hold K+2,K+3

  v8f acc0 = {};
  v8f acc4 = {};

  for (int j0 = 0; j0 < IN_DIM; j0 += JT) {
    __syncthreads();

    // ---- stage X tile: 128 b x 16 j, components {0,1,2,4} ----
    // 2048 (b,j) pairs / 256 threads = 8 each; consecutive threads -> 32B stride (coalesced-ish, L2 resident)
#pragma unroll
    for (int it = 0; it < 8; ++it) {
      int p  = t + it * 256;
      int bb = p >> 4;   // 0..127
      int jj = p & 15;   // 0..15
      const float* src = x + ((size_t)(bm + bb) * IN_DIM + (j0 + jj)) * MV;
      v4f v03 = *(const v4f*)src;  // comps 0..3 (32B aligned)
      float x4 = src[4];
      XSa[bb][jj * 3 + 0] = v03.x;
      XSa[bb][jj * 3 + 1] = v03.y;
      XSa[bb][jj * 3 + 2] = v03.z;
      XSb[bb][jj * 3 + 0] = v03.y;
      XSb[bb][jj * 3 + 1] = -v03.z;   // fold the minus sign into staging
      XSb[bb][jj * 3 + 2] = x4;
    }

    // ---- stage W tile: 16 n x 48 K, two packings -> 1536 floats, 6 per thread ----
#pragma unroll
    for (int it = 0; it < 6; ++it) {
      int q     = t + it * 256;     // 0..1535
      int nn    = q & 15;
      int K     = (q >> 4) & 63;    // 0..47 for first copy; &63 wraps second to 48..63, 0..31 before -48
      int which = (q >= 768);
      if (which) K -= 48;
      int jj = K / 3;
      int c  = K % 3;
      const float* wb = w + ((size_t)(j0 + jj) * OUT_DIM + (n0 + nn)) * MV;
      if (!which)
        WSa[nn][K] = wb[c];         // w0,w1,w2
      else
        WSb[nn][K] = wb[2 - c];     // w2,w1,w0
    }

    // prefetch a slice of the next x tile into cache
    if (j0 + JT < IN_DIM) {
      const float* nsrc =
          x + ((size_t)(bm + (t >> 4)) * IN_DIM + (j0 + JT + (t & 15))) * MV;
      __builtin_prefetch(nsrc, 0, 1);
    }

    __syncthreads();

    // ---- WMMA: wave wv owns batch rows [wv*16, wv*16+16), all 16 n's ----
    const int arow = wv * 16 + mhalf;  // A fragment row (M = lane&15 per ISA layout)
#pragma unroll
    for (int k0 = 0; k0 < KT; k0 += 4) {
      v2f a0 = *(const v2f*)&XSa[arow][k0 + koff];
      v2f b0 = *(const v2f*)&WSa[mhalf][k0 + koff];
      acc0 = __builtin_amdgcn_wmma_f32_16x16x4_f32(false, a0, false, b0,
                                                   (short)0, acc0, false, false);
      v2f a4 = *(const v2f*)&XSb[arow][k0 + koff];
      v2f b4 = *(const v2f*)&WSb[mhalf][k0 + koff];
      acc4 = __builtin_amdgcn_wmma_f32_16x16x4_f32(false, a4, false, b4,
                                                   (short)0, acc4, false, false);
    }
  }

  // ---- writeback: components 0 and 4 only (rest already hold bias) ----
  // D layout: VGPR r -> M=r (lanes 0-15) / M=8+r (lanes 16-31); N = lane&15
  const int ncol = n0 + mhalf;
  const float b0v = bias[ncol * MV + 0];
  const float b4v = bias[ncol * MV + 4];
  const int mbase = bm + wv * 16 + ((lane >> 4) ? 8 : 0);
#pragma unroll
  for (int r = 0; r < 8; ++r) {
    float* dst = y + ((size_t)(mbase + r) * OUT_DIM + ncol) * MV;
    dst[0] = acc0[r] + b0v;
    dst[4] = acc4[r] + b4v;
  }
}

extern "C" void kernel_launch(void* const* d_in, const int* in_sizes, int n_in,
                              void* d_out, int out_size, void* d_ws, size_t ws_size,
                              hipStream_t stream) {
  (void)in_sizes; (void)n_in; (void)d_ws; (void)ws_size; (void)out_size;
  const float* x    = (const float*)d_in[0];
  const float* wgt  = (const float*)d_in[1];
  const float* bias = (const float*)d_in[2];
  float* y = (float*)d_out;

  // Pass 1: bias broadcast (pure bandwidth; partial writes merge with pass 2 in the 192MB L2)
  const int total4 = B_DIM * OUT_DIM * MV / 4;  // 4,194,304 float4
  ga_fill_bias<<<2048, 256, 0, stream>>>((const float4*)bias, (float4*)y, total4);

  // Pass 2: WMMA GEMM for multivector components 0 and 4
  dim3 grid(B_DIM / MT, OUT_DIM / NT);  // (64, 16)
  ga_gemm<<<grid, 256, 0, stream>>>(x, wgt, bias, y);
}